// Transformer_74783970558259
// MI455X (gfx1250) — compile-verified
//
#include <hip/hip_runtime.h>
#include <stdint.h>

#define VOCAB 50257
#define SEQ   2048
#define EMB   512
#define BATCH 8

typedef float f4 __attribute__((ext_vector_type(4)));

// One block per sequence position s.
// 128 threads * float4 = 512 floats = one embedding row; loop over 8 batches.
__global__ __launch_bounds__(128) void emb_gather_add_kernel(
    const int*   __restrict__ tokens,  // (B, S) int32
    const float* __restrict__ W_emb,   // (E, V) row-major
    const float* __restrict__ W_pos,   // (S, E) row-major
    float*       __restrict__ out)     // (B, S, E)
{
    __shared__ float pos_row[EMB];

    const int s   = blockIdx.x;        // 0..2047
    const int tid = threadIdx.x;       // 0..127
    const int e0  = tid * 4;           // first of 4 consecutive embedding dims

    // ---- CDNA5 async data-mover: W_pos[s, :] -> LDS, 16B per lane ----
    {
        const float* gsrc   = W_pos + (size_t)s * EMB + e0;
        uint32_t lds_off    = (uint32_t)(uintptr_t)(&pos_row[e0]); // low 32b of flat ptr == LDS offset
        uint64_t gaddr      = (uint64_t)(uintptr_t)gsrc;
        asm volatile("global_load_async_to_lds_b128 %0, %1, off"
                     :: "v"(lds_off), "v"(gaddr)
                     : "memory");
    }

    // Block-uniform token ids for all batches at this position (scalar loads).
    int tok[BATCH];
#pragma unroll
    for (int b = 0; b < BATCH; ++b)
        tok[b] = tokens[b * SEQ + s];

    // Gather 4 strided floats per batch from column tok[b] of W_emb.
    // 32 independent loads in flight per thread -> good latency hiding;
    // W_emb (103MB) becomes L2-resident (192MB L2) so these mostly hit L2.
    float g[BATCH][4];
#pragma unroll
    for (int b = 0; b < BATCH; ++b) {
        const float* col = W_emb + (size_t)tok[b];
#pragma unroll
        for (int k = 0; k < 4; ++k)
            g[b][k] = col[(size_t)(e0 + k) * VOCAB];
    }

    // Wait for this wave's async LDS fill, then read back our own 16B chunk.
    asm volatile("s_wait_asynccnt 0" ::: "memory");
    f4 pos = *(const f4*)&pos_row[e0];

    // Add positional row and store with non-temporal hint (streaming output,
    // keep L2 for the embedding table).
#pragma unroll
    for (int b = 0; b < BATCH; ++b) {
        f4 v;
        v.x = g[b][0] + pos.x;
        v.y = g[b][1] + pos.y;
        v.z = g[b][2] + pos.z;
        v.w = g[b][3] + pos.w;
        f4* dst = (f4*)(out + ((size_t)b * SEQ + s) * EMB + e0);
        __builtin_nontemporal_store(v, dst);
    }
}

extern "C" void kernel_launch(void* const* d_in, const int* in_sizes, int n_in,
                              void* d_out, int out_size, void* d_ws, size_t ws_size,
                              hipStream_t stream) {
    const int*   tokens = (const int*)  d_in[0];   // (8, 2048) int32
    const float* W_emb  = (const float*)d_in[1];   // (512, 50257) f32
    const float* W_pos  = (const float*)d_in[2];   // (2048, 512) f32
    float*       out    = (float*)      d_out;     // (8, 2048, 512) f32

    dim3 grid(SEQ);
    dim3 block(128);
    emb_gather_add_kernel<<<grid, block, 0, stream>>>(tokens, W_emb, W_pos, out);
}